// GNN_77644418777510
// MI455X (gfx1250) — compile-verified
//
#include <hip/hip_runtime.h>
#include <hip/hip_bf16.h>

typedef __attribute__((ext_vector_type(2))) float v2f;
typedef __attribute__((ext_vector_type(8))) float v8f;

#define K_DIM 128  // both layers contract over 128 features

__device__ __forceinline__ void atomic_add_f32(float* p, float v) {
  __hip_atomic_fetch_add(p, v, __ATOMIC_RELAXED, __HIP_MEMORY_SCOPE_AGENT);
}

// ---- degree / rsqrt normalization -----------------------------------------
__global__ void deg_init_kernel(float* __restrict__ deg, int n) {
  int i = blockIdx.x * blockDim.x + threadIdx.x;
  if (i < n) deg[i] = 1.0f;  // self-loop weight contributes 1.0 to every node
}

__global__ void deg_acc_kernel(const long long* __restrict__ dst,
                               const float* __restrict__ ew,
                               float* __restrict__ deg, int e) {
  int i = blockIdx.x * blockDim.x + threadIdx.x;
  if (i < e) atomic_add_f32(&deg[dst[i]], ew[i]);
}

__global__ void deg_rsqrt_kernel(float* __restrict__ deg, int n) {
  int i = blockIdx.x * blockDim.x + threadIdx.x;
  if (i < n) {
    float d = deg[i];
    deg[i] = d > 0.0f ? rsqrtf(d) : 0.0f;  // becomes dis[i]
  }
}

// ---- fp32 WMMA GEMM: C[M x NC] = A[M x 128] @ W[128 x NC] -----------------
// One wave per 16x16 output tile; K loop in steps of 4 using
// V_WMMA_F32_16X16X4_F32 (exact fp32, matches reference precision).
template <int NC, bool RELU>
__global__ void gemm_wmma_kernel(const float* __restrict__ A,
                                 const float* __restrict__ W,
                                 float* __restrict__ C, int mtiles) {
  const int lane = threadIdx.x & 31;
  const int wave = threadIdx.x >> 5;
  constexpr int NT = NC / 16;
  int tile = blockIdx.x * (blockDim.x >> 5) + wave;
  if (tile >= mtiles * NT) return;  // wave-uniform guard: EXEC stays all-1s
  int mt = tile / NT;
  int nt = tile - mt * NT;
  int m0 = mt * 16, n0 = nt * 16;
  int half = lane >> 4;  // 0: lanes 0-15, 1: lanes 16-31
  int l = lane & 15;

  const float* arow = A + (size_t)(m0 + l) * K_DIM;  // A row for this lane
  const float* wcol = W + n0 + l;                    // B column for this lane
  v8f acc = {};
#pragma unroll 4
  for (int k = 0; k < K_DIM; k += 4) {
    int kk = k + half * 2;
    // A 16x4 fragment: lanes 0-15 hold K=k,k+1 ; lanes 16-31 hold K=k+2,k+3
    v2f a, b;
    a.x = arow[kk];
    a.y = arow[kk + 1];
    if (RELU) {
      a.x = fmaxf(a.x, 0.0f);
      a.y = fmaxf(a.y, 0.0f);
    }
    // B 4x16 fragment: VGPR v, lanes 0-15 -> row K=kk+v ; lanes 16-31 -> K=kk+v (kk already offset)
    b.x = wcol[(size_t)kk * NC];
    b.y = wcol[(size_t)(kk + 1) * NC];
    acc = __builtin_amdgcn_wmma_f32_16x16x4_f32(false, a, false, b,
                                                (short)0, acc, false, false);
  }
  // D 16x16: VGPR v -> row m0+v (lanes 0-15) / m0+v+8 (lanes 16-31), col n0+l
  float* crow = C + (size_t)(m0 + half * 8) * NC + n0 + l;
#pragma unroll
  for (int v = 0; v < 8; ++v) crow[(size_t)v * NC] = acc[v];
}

// ---- out[i][c] = bias[c] + dis[i]^2 * h[i][c]  (self-loop term + bias) ----
template <int CH>
__global__ void self_bias_init_kernel(const float* __restrict__ h,
                                      const float* __restrict__ bias,
                                      const float* __restrict__ dis,
                                      float* __restrict__ out, int n) {
  long long idx = (long long)blockIdx.x * blockDim.x + threadIdx.x;
  if (idx < (long long)n * CH) {
    int i = (int)(idx / CH);
    int c = (int)(idx - (long long)i * CH);
    float s = dis[i];
    out[idx] = bias[c] + h[idx] * (s * s);
  }
}

// ---- edge scatter: one wave per edge, VEC channels per lane ---------------
template <int CH>
__global__ void edge_scatter_kernel(const float* __restrict__ h,
                                    const long long* __restrict__ src,
                                    const long long* __restrict__ dst,
                                    const float* __restrict__ ew,
                                    const float* __restrict__ dis,
                                    float* __restrict__ out, int ne) {
  const int lane = threadIdx.x & 31;
  int e = blockIdx.x * (blockDim.x >> 5) + (threadIdx.x >> 5);
  if (e >= ne) return;  // wave-uniform guard
  long long s = src[e];  // wave-uniform -> scalar loads
  long long d = dst[e];
  float norm = dis[s] * ew[e] * dis[d];
  constexpr int VEC = CH / 32;
  const float* hp = h + (size_t)s * CH + lane * VEC;
  float* op = out + (size_t)d * CH + lane * VEC;
  float vals[VEC];
#pragma unroll
  for (int v = 0; v < VEC; ++v) vals[v] = hp[v];  // contiguous -> b64/b128 load
#pragma unroll
  for (int v = 0; v < VEC; ++v) atomic_add_f32(op + v, vals[v] * norm);
}

extern "C" void kernel_launch(void* const* d_in, const int* in_sizes, int n_in,
                              void* d_out, int out_size, void* d_ws, size_t ws_size,
                              hipStream_t stream) {
  (void)n_in; (void)out_size; (void)ws_size;
  const float* x = (const float*)d_in[0];
  const long long* ei = (const long long*)d_in[1];
  const float* ew = (const float*)d_in[2];
  const float* W1 = (const float*)d_in[3];
  const float* b1 = (const float*)d_in[4];
  const float* W2 = (const float*)d_in[5];
  const float* b2 = (const float*)d_in[6];

  const int IN_C = 128, HID_C = 128, OUT_C = 64;
  const int N = in_sizes[0] / IN_C;  // 100000
  const int E = in_sizes[2];         // 1600000
  const long long* src = ei;
  const long long* dst = ei + E;

  // workspace carve-out (256B aligned): dis(N) | h1(N*128) | out1(N*128)
  char* ws = (char*)d_ws;
  size_t off = 0;
  auto carve = [&](size_t bytes) -> char* {
    char* p = ws + off;
    off += (bytes + 255) & ~(size_t)255;
    return p;
  };
  float* dis  = (float*)carve((size_t)N * sizeof(float));
  float* h1   = (float*)carve((size_t)N * HID_C * sizeof(float));
  float* out1 = (float*)carve((size_t)N * HID_C * sizeof(float));
  float* h2   = h1;  // h1 is dead once edge_scatter<128> completes (stream-ordered)
  float* out  = (float*)d_out;

  const int T = 256;                 // 8 waves per block on wave32
  const int WPB = T / 32;

  // 1) symmetric normalization coefficients dis[i] = rsqrt(deg[i])
  deg_init_kernel<<<(N + T - 1) / T, T, 0, stream>>>(dis, N);
  deg_acc_kernel<<<(E + T - 1) / T, T, 0, stream>>>(dst, ew, dis, E);
  deg_rsqrt_kernel<<<(N + T - 1) / T, T, 0, stream>>>(dis, N);

  const int mtiles = N / 16;  // 6250, exact

  // 2) layer 1: h1 = x @ W1 (WMMA fp32)
  {
    int tiles = mtiles * (HID_C / 16);
    gemm_wmma_kernel<HID_C, false>
        <<<(tiles + WPB - 1) / WPB, T, 0, stream>>>(x, W1, h1, mtiles);
  }
  // 3) out1 = b1 + dis^2*h1 (self loop), then scatter edges
  self_bias_init_kernel<HID_C>
      <<<(int)(((long long)N * HID_C + T - 1) / T), T, 0, stream>>>(h1, b1, dis, out1, N);
  edge_scatter_kernel<HID_C>
      <<<(E + WPB - 1) / WPB, T, 0, stream>>>(h1, src, dst, ew, dis, out1, E);

  // 4) layer 2: h2 = relu(out1) @ W2 (ReLU fused into A loads)
  {
    int tiles = mtiles * (OUT_C / 16);
    gemm_wmma_kernel<OUT_C, true>
        <<<(tiles + WPB - 1) / WPB, T, 0, stream>>>(out1, W2, h2, mtiles);
  }
  // 5) out = b2 + dis^2*h2, then scatter edges
  self_bias_init_kernel<OUT_C>
      <<<(int)(((long long)N * OUT_C + T - 1) / T), T, 0, stream>>>(h2, b2, dis, out, N);
  edge_scatter_kernel<OUT_C>
      <<<(E + WPB - 1) / WPB, T, 0, stream>>>(h2, src, dst, ew, dis, out, E);
}